// SAGENet_38697655336972
// MI455X (gfx1250) — compile-verified
//
#include <hip/hip_runtime.h>

#define N_NODES 100000
#define N_EDGES 1600000
#define IN_DIM  128
#define HID_DIM 128
#define OUT_DIM 64

typedef __attribute__((ext_vector_type(2))) float v2f;
typedef __attribute__((ext_vector_type(8))) float v8f;

// -------------------------------------------------------------------------
// Edge scatter: one wave32 per edge; each lane carries 4 of the 128 features.
// agg[dst] += feat[src]  (f32 atomics, accumulator is L2-resident: 51MB<192MB)
// deg[dst] += 1 (lane 0, first layer only)
// -------------------------------------------------------------------------
__global__ void __launch_bounds__(256)
scatter_sum_kernel(const float* __restrict__ feat,
                   const int*   __restrict__ src,
                   const int*   __restrict__ dst,
                   float*       __restrict__ agg,
                   float*       __restrict__ deg,
                   int n_edges) {
  int gid  = blockIdx.x * 256 + threadIdx.x;
  int lane = gid & 31;
  int edge = gid >> 5;
  if (edge >= n_edges) return;
  int s = src[edge];
  int d = dst[edge];
  float4 v = ((const float4*)(feat + (size_t)s * 128))[lane];
  float* o = agg + (size_t)d * 128 + lane * 4;
  atomicAdd(o + 0, v.x);
  atomicAdd(o + 1, v.y);
  atomicAdd(o + 2, v.z);
  atomicAdd(o + 3, v.w);
  if (deg != nullptr && lane == 0) atomicAdd(deg + d, 1.0f);
}

// -------------------------------------------------------------------------
// Fused SAGE linear:  out = [relu]( (agg/deg) @ Wl^T + x @ Wr^T + b )
// One block per 16-node tile. Block stages both 16x128 A-tiles into LDS
// (coalesced float4, row stride 132 floats -> conflict-free b64 reads).
// Wave w computes output columns [16w, 16w+16) with V_WMMA_F32_16X16X4_F32,
// K=128 -> 32 WMMA steps per A operand, 64 per wave total, fp32 end to end.
//   A layout (16x4 f32): lane l -> row l%16, holds K = kbase+2*(l/16)+{0,1}
//   B layout (4x16 f32): lane l -> col l%16, holds K = kbase+2*(l/16)+{0,1}
//     and B[k][n] = W[n][k], so each lane reads a contiguous float2 of W row n.
//   C/D layout: lane l -> n=l%16 ; vgpr v -> m = v + 8*(l/16)
// -------------------------------------------------------------------------
template <int NOUT, bool RELU>
__global__ void __launch_bounds__(NOUT * 2)
sage_linear_kernel(const float* __restrict__ agg,
                   const float* __restrict__ deg,
                   const float* __restrict__ xin,
                   const float* __restrict__ Wl,    // [NOUT][128] row-major
                   const float* __restrict__ Wr,    // [NOUT][128] row-major
                   const float* __restrict__ bias,  // [NOUT]
                   float*       __restrict__ out) { // [N_NODES][NOUT]
  constexpr int K      = 128;
  constexpr int LSTR   = K + 4;      // 132 floats: conflict-free LDS reads
  constexpr int NWAVES = NOUT / 16;

  __shared__ float lds_a[16 * LSTR];
  __shared__ float lds_x[16 * LSTR];

  const int tid = threadIdx.x;
  const int m0  = blockIdx.x * 16;

  // Cooperative, coalesced staging of the two A tiles (agg pre-scaled by 1/deg).
  const float4* ga = (const float4*)(agg + (size_t)m0 * K);
  const float4* gx = (const float4*)(xin + (size_t)m0 * K);
  for (int i = tid; i < 16 * (K / 4); i += NWAVES * 32) {
    int row = i >> 5;          // 32 float4 per row
    int c4  = i & 31;
    float invd = 1.0f / fmaxf(deg[m0 + row], 1.0f);
    float4 a = ga[i];
    a.x *= invd; a.y *= invd; a.z *= invd; a.w *= invd;
    *(float4*)(lds_a + row * LSTR + c4 * 4) = a;
    *(float4*)(lds_x + row * LSTR + c4 * 4) = gx[i];
  }
  __syncthreads();

  const int lane  = tid & 31;
  const int wave  = tid >> 5;
  const int nb    = wave * 16;      // output-column base for this wave
  const int mlane = lane & 15;
  const int kh    = (lane >> 4) << 1;  // 0 or 2

  const float* wl = Wl + (size_t)(nb + mlane) * K + kh;
  const float* wr = Wr + (size_t)(nb + mlane) * K + kh;
  const float* la = lds_a + mlane * LSTR + kh;
  const float* lx = lds_x + mlane * LSTR + kh;

  v8f c = {};
#pragma unroll
  for (int ks = 0; ks < K; ks += 4) {
    v2f a = *(const v2f*)(la + ks);
    v2f b = *(const v2f*)(wl + ks);
    c = __builtin_amdgcn_wmma_f32_16x16x4_f32(false, a, false, b, (short)0, c,
                                              false, false);
  }
#pragma unroll
  for (int ks = 0; ks < K; ks += 4) {
    v2f a = *(const v2f*)(lx + ks);
    v2f b = *(const v2f*)(wr + ks);
    c = __builtin_amdgcn_wmma_f32_16x16x4_f32(false, a, false, b, (short)0, c,
                                              false, false);
  }

  const float bv = bias[nb + mlane];
  const int   mh = (lane >> 4) << 3;  // 0 or 8
#pragma unroll
  for (int v = 0; v < 8; v++) {
    float r = c[v] + bv;
    if (RELU) r = fmaxf(r, 0.0f);
    out[(size_t)(m0 + v + mh) * NOUT + nb + mlane] = r;
  }
}

// -------------------------------------------------------------------------
extern "C" void kernel_launch(void* const* d_in, const int* in_sizes, int n_in,
                              void* d_out, int out_size, void* d_ws,
                              size_t ws_size, hipStream_t stream) {
  const float* x   = (const float*)d_in[0];
  const int*   ei  = (const int*)d_in[1];   // [2, N_EDGES] int32
  const float* W1l = (const float*)d_in[2];
  const float* W1r = (const float*)d_in[3];
  const float* b1  = (const float*)d_in[4];
  const float* W2l = (const float*)d_in[5];
  const float* W2r = (const float*)d_in[6];
  const float* b2  = (const float*)d_in[7];
  float*       out = (float*)d_out;

  const int* src = ei;
  const int* dst = ei + N_EDGES;

  float* ws   = (float*)d_ws;
  float* agg1 = ws;                              // N*128
  float* h    = agg1 + (size_t)N_NODES * 128;    // N*128
  float* agg2 = h    + (size_t)N_NODES * 128;    // N*128
  float* deg  = agg2 + (size_t)N_NODES * 128;    // N

  hipMemsetAsync(agg1, 0, (size_t)N_NODES * 128 * sizeof(float), stream);
  hipMemsetAsync(agg2, 0, (size_t)N_NODES * 128 * sizeof(float), stream);
  hipMemsetAsync(deg,  0, (size_t)N_NODES * sizeof(float), stream);

  const dim3 sgrid(N_EDGES / 8);  // 8 edges (waves) per 256-thread block

  // Layer 1: aggregate x, then h = relu((agg1/deg)@W1l^T + x@W1r^T + b1)
  scatter_sum_kernel<<<sgrid, 256, 0, stream>>>(x, src, dst, agg1, deg,
                                                N_EDGES);
  sage_linear_kernel<HID_DIM, true><<<N_NODES / 16, 256, 0, stream>>>(
      agg1, deg, x, W1l, W1r, b1, h);

  // Layer 2: aggregate h (deg unchanged), out = (agg2/deg)@W2l^T + h@W2r^T + b2
  scatter_sum_kernel<<<sgrid, 256, 0, stream>>>(h, src, dst, agg2, nullptr,
                                                N_EDGES);
  sage_linear_kernel<OUT_DIM, false><<<N_NODES / 16, 128, 0, stream>>>(
      agg2, deg, h, W2l, W2r, b2, out);
}